// EvalBspPrimeTF_18580028523045
// MI455X (gfx1250) — compile-verified
//
#include <hip/hip_runtime.h>

typedef float v2f __attribute__((ext_vector_type(2)));
typedef float v4f __attribute__((ext_vector_type(4)));
typedef float v8f __attribute__((ext_vector_type(8)));

#define NCOL 17             // ORDER+1 output columns
#define PTS_PER_BLOCK 256
#define THREADS 256

// out[n,m] = 16*(B15[m-1](y) - B15[m](y)),  B15[k] = binom(15,k) y^k (1-y)^(15-k)
// Realized as D = A x T with 4 chained V_WMMA_F32_16X16X4_F32 per 16-point tile.
// A fragment per lane: rows = point (lane&15), K pair = 4s + 2*hi + {0,1}.
__global__ __launch_bounds__(THREADS)
void bsp_prime_wmma_kernel(const float* __restrict__ x,
                           float* __restrict__ out,
                           int npts) {
  __shared__ float lds[PTS_PER_BLOCK * NCOL];   // 17408 B staging

  const int tid  = threadIdx.x;
  const int lane = tid & 31;
  const int wave = tid >> 5;
  const int n    = lane & 15;      // D column / point-in-tile index
  const int hi   = lane >> 4;      // half-wave selects K pair {2,3} vs {0,1}
  const int d    = n - 2 * hi;     // column index folded with half-wave K offset

  // ---- one-time per-lane constants (kept live across the tile loop) ----
  // B fragment of T (difference stencil): row k = 4s+2hi (+1), column n.
  // T[k,m] = -16 at m==k, +16 at m==k+1   ==>   vs d: literal-only compares.
  v2f tb[4];
#pragma unroll
  for (int s = 0; s < 4; ++s) {
    tb[s].x = (d == 4 * s)     ? -16.0f : ((d == 4 * s + 1) ? 16.0f : 0.0f);
    tb[s].y = (d == 4 * s + 1) ? -16.0f : ((d == 4 * s + 2) ? 16.0f : 0.0f);
  }
  const float C15[16] = {1.f, 15.f, 105.f, 455.f, 1365.f, 3003.f, 5005.f, 6435.f,
                         6435.f, 5005.f, 3003.f, 1365.f, 455.f, 105.f, 15.f, 1.f};
  float cA[4], cB[4];              // binom(15, 4s+2hi), binom(15, 4s+2hi+1)
#pragma unroll
  for (int s = 0; s < 4; ++s) {
    cA[s] = hi ? C15[4 * s + 2] : C15[4 * s + 0];
    cB[s] = hi ? C15[4 * s + 3] : C15[4 * s + 1];
  }

  const int blockBase = blockIdx.x * PTS_PER_BLOCK;

#pragma unroll 1                                  // real loop: LICM hoists tb/cA/cB
  for (int t = 0; t < 2; ++t) {                   // 2 tiles of 16 points per wave
    const int tileP0 = wave * 32 + t * 16;        // point offset within block
    int gidx = blockBase + tileP0 + n;
    if (gidx >= npts) gidx = npts - 1;            // clamp; stores guarded below
    const float y = x[gidx];
    const float u = 1.0f - y;
    // Reference's nan_to_zero: exact y==0 / y==1 yields an all-zero row.
    const float valid = (y > 0.0f && y < 1.0f) ? 1.0f : 0.0f;

    const float y2 = y * y, y4 = y2 * y2, y8 = y4 * y4, y12 = y8 * y4;
    const float u2 = u * u, u3 = u2 * u, u4 = u2 * u2, u8 = u4 * u4, u12 = u8 * u4;

    const float wy  = valid * (hi ? y2 : 1.0f);   // valid * y^(2hi)
    const float wuA = hi ? u : u3;                // u^(3-2hi)
    const float wuB = hi ? 1.0f : u2;             // u^(2-2hi)
    const float Y4s[4]  = {1.0f, y4, y8, y12};    // y^(4s)
    const float U12s[4] = {u12, u8, u4, 1.0f};    // u^(12-4s)

    v8f acc = {};
#pragma unroll
    for (int s = 0; s < 4; ++s) {
      const float t1 = Y4s[s] * wy;               // valid * y^(4s+2hi)
      v2f a;
      a.x = (cA[s] * t1)       * (U12s[s] * wuA); // valid*B15[4s+2hi]
      a.y = (cB[s] * (t1 * y)) * (U12s[s] * wuB); // valid*B15[4s+2hi+1]
      acc = __builtin_amdgcn_wmma_f32_16x16x4_f32(false, a, false, tb[s],
                                                  (short)0, acc, false, false);
    }

    // D tile: lane holds column n, rows M = v + 8*hi -> row-major LDS staging.
    const int base = (tileP0 + 8 * hi) * NCOL + n;
#pragma unroll
    for (int v = 0; v < 8; ++v) lds[base + v * NCOL] = acc[v];
    // 17th column: 16 * B15[15] = 16 * y^15 (masked).
    if (!hi) lds[(tileP0 + n) * NCOL + 16] = (16.0f * valid) * (y12 * y2 * y);
  }

  __syncthreads();

  // Coalesced streaming store: 256*17 dwords = 1088 float4 per block.
  if (blockBase + PTS_PER_BLOCK <= npts) {
    const v4f* lds4 = (const v4f*)lds;
    v4f* out4 = (v4f*)(out + (size_t)blockBase * NCOL);   // 16B-aligned: 256*68
#pragma unroll
    for (int i = 0; i < 4; ++i)
      __builtin_nontemporal_store(lds4[i * THREADS + tid], &out4[i * THREADS + tid]);
    if (tid < 64)
      __builtin_nontemporal_store(lds4[4 * THREADS + tid], &out4[4 * THREADS + tid]);
  } else {                                        // guarded tail path
    const size_t total = (size_t)npts * NCOL;
    const size_t ob    = (size_t)blockBase * NCOL;
#pragma unroll 1
    for (int i = 0; i < NCOL; ++i) {
      const size_t idx = ob + (size_t)i * THREADS + tid;
      if (idx < total) out[idx] = lds[i * THREADS + tid];
    }
  }
}

extern "C" void kernel_launch(void* const* d_in, const int* in_sizes, int n_in,
                              void* d_out, int out_size, void* d_ws, size_t ws_size,
                              hipStream_t stream) {
  const float* x = (const float*)d_in[0];
  float* out = (float*)d_out;
  const int npts = in_sizes[0];                                   // 4,000,000
  const int blocks = (npts + PTS_PER_BLOCK - 1) / PTS_PER_BLOCK;  // 15625
  bsp_prime_wmma_kernel<<<blocks, THREADS, 0, stream>>>(x, out, npts);
}